// VectorQuantizer_30013231464711
// MI455X (gfx1250) — compile-verified
//
#include <hip/hip_runtime.h>
#include <hip/hip_bf16.h>

typedef __attribute__((ext_vector_type(2))) float v2f;
typedef __attribute__((ext_vector_type(8))) float v8f;

#define DIM        64
#define KCODES     512
#define PSTRIDE    528              // pair-row stride in float2s: 2*528*2 mod 64 banks = 32
#define NROWS      262144
#define ROWS_PER_BLOCK 512
#define NBLK       512
#define QELEMS     16777216         // 64^4 quantized floats
#define LOSS_OFF   16777216
#define PERP_OFF   16777217
#define IDX_OFF    16777218

// ---------------- init: codebook norms + zero histogram ----------------
__global__ void vq_init(const float* __restrict__ emb, float* __restrict__ enorm,
                        unsigned* __restrict__ counts) {
  const int k = threadIdx.x;                 // 512 threads, one per code
  float s = 0.0f;
  #pragma unroll 8
  for (int d = 0; d < DIM; ++d) {
    const float e = emb[d * KCODES + k];
    s = fmaf(e, e, s);
  }
  enorm[k] = s;
  counts[k] = 0u;
}

// ---------------- main: WMMA scores -> argmin -> gather/write ----------------
__launch_bounds__(256, 1)
__global__ void vq_main(const float* __restrict__ x, const float* __restrict__ emb,
                        const float* __restrict__ enorm, unsigned* __restrict__ counts,
                        float* __restrict__ partials, float* __restrict__ outq,
                        float* __restrict__ outidx) {
  // Codebook in LDS as interleaved row pairs: El2[p*PSTRIDE + c] = {E[2p][c], E[2p+1][c]}
  // -> each WMMA B-fragment is ONE ds_load_b64, bank-conflict-free across the wave.
  __shared__ __align__(16) float2 El2[32 * PSTRIDE];    // 135168 B
  __shared__ float enorm_s[KCODES];                     // 2 KB
  __shared__ __align__(16) float qstage[8 * 16 * 68];   // 34816 B per-wave gather staging
  __shared__ int   idx_s[8 * 16];
  __shared__ float red_s[8];

  const int tid  = threadIdx.x;
  const int wave = tid >> 5;
  const int lane = tid & 31;
  const int lh   = lane & 15;       // position within half-wave
  const int hi   = lane >> 4;       // half select (K-split for A/B fragments)

  // Stage codebook into LDS as row pairs (coalesced global reads, conflict-free b64 stores)
  {
    for (int i = tid; i < 32 * KCODES; i += 256) {
      const int p = i >> 9;         // pair index 0..31 (rows 2p, 2p+1)
      const int c = i & 511;
      float2 v;
      v.x = emb[(size_t)(2 * p) * KCODES + c];
      v.y = emb[(size_t)(2 * p + 1) * KCODES + c];
      El2[p * PSTRIDE + c] = v;
    }
    for (int i = tid; i < KCODES; i += 256) enorm_s[i] = enorm[i];
  }
  __syncthreads();

  float sumsq = 0.0f;

  for (int strip = 0; strip < 4; ++strip) {
    const int row_base = blockIdx.x * ROWS_PER_BLOCK + strip * 128 + wave * 16;

    // Prefetch next strip's A rows (16 rows x 256B = 4KB contiguous; 32 lanes x 128B)
    if (strip < 3) {
      const char* nx = (const char*)(x + (size_t)(row_base + 128) * DIM) + lane * 128;
      __builtin_prefetch(nx, 0, 0);
    }

    // A fragments: 16 chunks of 16x4 f32. ISA layout: lanes 0-15 hold K={4j,4j+1},
    // lanes 16-31 hold K={4j+2,4j+3}; M = lane&15.
    v2f a[16];
    {
      const float* xr = x + (size_t)(row_base + lh) * DIM + 2 * hi;
      #pragma unroll
      for (int j = 0; j < 16; ++j) {
        const float2 t = *(const float2*)(xr + 4 * j);
        a[j].x = t.x;
        a[j].y = t.y;
      }
    }

    float best[8];
    int   bidx[8];
    #pragma unroll
    for (int r = 0; r < 8; ++r) { best[r] = 3.0e38f; bidx[r] = 0x7fffffff; }

    for (int ct = 0; ct < 32; ++ct) {          // 32 column tiles of 16 codes
      const int cbase = ct * 16 + lh;
      const float2* Bp = El2 + cbase;
      // Two independent accumulator chains (K 0..31 and K 32..63) for WMMA ILP
      v8f acc0 = {}, acc1 = {};
      #pragma unroll
      for (int j = 0; j < 8; ++j) {
        const int p0 = 2 * j + hi;             // pair holding K = {4j+2hi, 4j+2hi+1}
        const int p1 = 2 * (j + 8) + hi;
        const float2 b0 = Bp[p0 * PSTRIDE];
        const float2 b1 = Bp[p1 * PSTRIDE];
        v2f vb0; vb0.x = b0.x; vb0.y = b0.y;
        v2f vb1; vb1.x = b1.x; vb1.y = b1.y;
        acc0 = __builtin_amdgcn_wmma_f32_16x16x4_f32(
            false, a[j],     false, vb0, (short)0, acc0, false, false);
        acc1 = __builtin_amdgcn_wmma_f32_16x16x4_f32(
            false, a[j + 8], false, vb1, (short)0, acc1, false, false);
      }
      // score = ||e||^2 - 2 x.e  (||x||^2 constant per row, irrelevant for argmin)
      const float en = enorm_s[cbase];
      #pragma unroll
      for (int r = 0; r < 8; ++r) {
        const float score = fmaf(-2.0f, acc0[r] + acc1[r], en);
        if (score < best[r]) { best[r] = score; bidx[r] = cbase; }
      }
    }

    // argmin across the 16 lanes of each half (ties -> lowest index, matching argmax-first)
    #pragma unroll
    for (int off = 1; off < 16; off <<= 1) {
      #pragma unroll
      for (int r = 0; r < 8; ++r) {
        const float os = __shfl_xor(best[r], off, 32);
        const int   oi = __shfl_xor(bidx[r], off, 32);
        if (os < best[r] || (os == best[r] && oi < bidx[r])) { best[r] = os; bidx[r] = oi; }
      }
    }

    // lane 0 owns rows row_base+0..7 ; lane 16 owns rows row_base+8..15
    if (lh == 0) {
      #pragma unroll
      for (int r = 0; r < 8; ++r) {
        const int gi = bidx[r];
        const int rr = hi * 8 + r;
        idx_s[wave * 16 + rr] = gi;
        outidx[row_base + rr] = (float)gi;
        atomicAdd(&counts[gi], 1u);            // integer atomic: deterministic
      }
    }
    __syncthreads();

    // Gather chosen codebook columns into staging (emb is L2/WGP$-resident: 128 KB)
    {
      const int   myidx = idx_s[wave * 16 + lh];
      const int   d0    = hi * 32;
      float* qs = qstage + wave * 1088 + lh * 68;
      #pragma unroll
      for (int dd = 0; dd < 32; ++dd)
        qs[d0 + dd] = emb[(size_t)(d0 + dd) * KCODES + myidx];
    }
    __syncthreads();

    // Drain staging: coalesced float4 stores of quantized + squared-error accumulation
    {
      const float4* x4 = ((const float4*)x) + (size_t)row_base * 16;
      float4*       o4 = ((float4*)outq)   + (size_t)row_base * 16;
      #pragma unroll
      for (int j2 = 0; j2 < 8; ++j2) {
        const int t  = j2 * 32 + lane;         // 0..255 float4s = 16 rows x 64 floats
        const int r  = t >> 4;
        const int c4 = t & 15;
        const float4 q  = ((const float4*)(qstage + wave * 1088 + r * 68))[c4];
        const float4 xv = x4[t];
        o4[t] = q;
        const float e0 = q.x - xv.x, e1 = q.y - xv.y, e2 = q.z - xv.z, e3 = q.w - xv.w;
        sumsq += e0 * e0 + e1 * e1 + e2 * e2 + e3 * e3;
      }
    }
    __syncthreads();
  }

  // Deterministic block reduction of squared error -> per-block partial
  #pragma unroll
  for (int off = 16; off >= 1; off >>= 1) sumsq += __shfl_xor(sumsq, off, 32);
  if (lane == 0) red_s[wave] = sumsq;
  __syncthreads();
  if (tid == 0) {
    float s = 0.0f;
    #pragma unroll
    for (int w = 0; w < 8; ++w) s += red_s[w];
    partials[blockIdx.x] = s;
  }
}

// ---------------- final: loss + perplexity ----------------
__global__ void vq_final(const unsigned* __restrict__ counts,
                         const float* __restrict__ partials,
                         float* __restrict__ out) {
  __shared__ float sb[32];
  const int tid = threadIdx.x;                 // 512 threads
  float ls = partials[tid];                    // NBLK == 512 partials
  const float p = (float)counts[tid] * (1.0f / (float)NROWS);
  float ent = p * logf(p + 1e-10f);
  #pragma unroll
  for (int off = 16; off >= 1; off >>= 1) {
    ls  += __shfl_xor(ls, off, 32);
    ent += __shfl_xor(ent, off, 32);
  }
  const int wave = tid >> 5;
  const int lane = tid & 31;
  if (lane == 0) { sb[wave] = ls; sb[16 + wave] = ent; }
  __syncthreads();
  if (tid == 0) {
    float L = 0.0f, E = 0.0f;
    for (int w = 0; w < 16; ++w) { L += sb[w]; E += sb[16 + w]; }
    // q_latent + commitment * e_latent == 1.25 * mean((q-x)^2) in forward pass
    out[LOSS_OFF] = 1.25f * (L * (1.0f / (float)QELEMS));
    out[PERP_OFF] = expf(-E);
  }
}

extern "C" void kernel_launch(void* const* d_in, const int* in_sizes, int n_in,
                              void* d_out, int out_size, void* d_ws, size_t ws_size,
                              hipStream_t stream) {
  (void)in_sizes; (void)n_in; (void)out_size; (void)ws_size;
  const float* x   = (const float*)d_in[0];   // [64,64,64,64] f32
  const float* emb = (const float*)d_in[1];   // [64,512] f32
  float* out = (float*)d_out;

  float*    enorm    = (float*)d_ws;                         // 512 f32
  unsigned* counts   = (unsigned*)((char*)d_ws + 2048);      // 512 u32
  float*    partials = (float*)((char*)d_ws + 4096);         // 512 f32

  vq_init <<<1, 512, 0, stream>>>(emb, enorm, counts);
  vq_main <<<NBLK, 256, 0, stream>>>(x, emb, enorm, counts, partials,
                                     out, out + IDX_OFF);
  vq_final<<<1, 512, 0, stream>>>(counts, partials, out);
}